// WordDecoderARNN_56178172231685
// MI455X (gfx1250) — compile-verified
//
#include <hip/hip_runtime.h>

#define BB   128
#define TT   2048
#define VDIM 128
#define ADIM 10
#define HH   64
#define HS1  96
#define HS2  48
#define NC   4000
#define STEPS 32

typedef __attribute__((ext_vector_type(2))) float v2f;
typedef __attribute__((ext_vector_type(8))) float v8f;

// -------- init: zero recurrent state + previous-attention buffer --------
__global__ void k_init(float* __restrict__ h0, float* __restrict__ h1,
                       float* __restrict__ aprev) {
    int i = blockIdx.x * blockDim.x + threadIdx.x;
    if (i < BB * HH) { h0[i] = 0.f; h1[i] = 0.f; }
    if (i < BB * TT) aprev[i] = 0.f;
}

// -------- v_proj = enc (B*T x 128) @ wv^T (128 x 10), padded N=16 --------
// One wave per 16-row tile, V_WMMA_F32_16X16X4_F32, K stepped by 4.
__global__ void k_vproj(const float* __restrict__ enc,
                        const float* __restrict__ wv,
                        float* __restrict__ vproj) {
    int gw   = (blockIdx.x * blockDim.x + threadIdx.x) >> 5;  // tile id
    int lane = threadIdx.x & 31;
    long row0 = (long)gw * 16;
    int m  = lane & 15;            // A row / D column
    int kh = (lane >> 4) * 2;      // K half-select (0 or 2)
    const float* arow = enc + (row0 + m) * VDIM + kh;
    int n    = lane & 15;          // B / D column
    int ncl  = (n < ADIM) ? n : 0;
    float msk = (n < ADIM) ? 1.f : 0.f;
    const float* brow = wv + ncl * VDIM + kh;
    v8f acc = {};
    for (int k0 = 0; k0 < VDIM; k0 += 4) {
        v2f a;  a.x = arow[k0];        a.y = arow[k0 + 1];
        v2f bf; bf.x = brow[k0] * msk; bf.y = brow[k0 + 1] * msk;
        acc = __builtin_amdgcn_wmma_f32_16x16x4_f32(
            false, a, false, bf, (short)0, acc, false, false);
    }
    int mr = (lane >> 4) * 8;  // D: vgpr r -> row r (lanes 0-15) / r+8 (16-31)
    float* dst = vproj + (row0 + mr) * 16 + n;
#pragma unroll
    for (int r = 0; r < 8; ++r) dst[r * 16] = acc[r];
}

// -------- 2-layer GRU step + q projection (x0 == 0 so gi0 = b_ih0) --------
__global__ void k_gru(const float* __restrict__ whh0, const float* __restrict__ bih0,
                      const float* __restrict__ bhh0,
                      const float* __restrict__ wih1, const float* __restrict__ whh1,
                      const float* __restrict__ bih1, const float* __restrict__ bhh1,
                      const float* __restrict__ wq,   const float* __restrict__ bq,
                      float* __restrict__ h0g, float* __restrict__ h1g,
                      float* __restrict__ qg) {
    __shared__ float h0s[16 * HH], h1s[16 * HH], h0n[16 * HH], h1n[16 * HH];
    int tid = threadIdx.x;
    int b0  = blockIdx.x * 16;
    for (int i = tid; i < 16 * HH; i += 256) {
        h0s[i] = h0g[b0 * HH + i];
        h1s[i] = h1g[b0 * HH + i];
    }
    __syncthreads();
    // layer 0 (input gates = b_ih0)
    for (int idx = tid; idx < 16 * HH; idx += 256) {
        int bb = idx >> 6, j = idx & 63;
        const float* hrow = h0s + bb * HH;
        float ar = 0.f, az = 0.f, an = 0.f;
        const float* wr = whh0 + j * HH;
        const float* wz = whh0 + (64 + j) * HH;
        const float* wn = whh0 + (128 + j) * HH;
        for (int k = 0; k < HH; ++k) {
            float h = hrow[k]; ar += h * wr[k]; az += h * wz[k]; an += h * wn[k];
        }
        float r = 1.f / (1.f + expf(-(bih0[j] + bhh0[j] + ar)));
        float z = 1.f / (1.f + expf(-(bih0[64 + j] + bhh0[64 + j] + az)));
        float n = tanhf(bih0[128 + j] + r * (an + bhh0[128 + j]));
        h0n[idx] = (1.f - z) * n + z * hrow[j];
    }
    __syncthreads();
    // layer 1
    for (int idx = tid; idx < 16 * HH; idx += 256) {
        int bb = idx >> 6, j = idx & 63;
        const float* xrow = h0n + bb * HH;
        const float* hrow = h1s + bb * HH;
        float ir = 0.f, iz = 0.f, inn = 0.f, hr = 0.f, hz = 0.f, hn = 0.f;
        const float* wir = wih1 + j * HH;         const float* whr = whh1 + j * HH;
        const float* wiz = wih1 + (64 + j) * HH;  const float* whz = whh1 + (64 + j) * HH;
        const float* win = wih1 + (128 + j) * HH; const float* whn = whh1 + (128 + j) * HH;
        for (int k = 0; k < HH; ++k) {
            float x = xrow[k], h = hrow[k];
            ir += x * wir[k]; iz += x * wiz[k]; inn += x * win[k];
            hr += h * whr[k]; hz += h * whz[k]; hn += h * whn[k];
        }
        float r = 1.f / (1.f + expf(-(ir + bih1[j] + hr + bhh1[j])));
        float z = 1.f / (1.f + expf(-(iz + bih1[64 + j] + hz + bhh1[64 + j])));
        float n = tanhf(inn + bih1[128 + j] + r * (hn + bhh1[128 + j]));
        h1n[idx] = (1.f - z) * n + z * hrow[j];
    }
    __syncthreads();
    for (int i = tid; i < 16 * HH; i += 256) {
        h0g[b0 * HH + i] = h0n[i];
        h1g[b0 * HH + i] = h1n[i];
    }
    // q = h1' @ wq^T + bq  (16 b x 10)
    for (int idx = tid; idx < 16 * ADIM; idx += 256) {
        int bb = idx / ADIM, a = idx % ADIM;
        const float* hrow = h1n + bb * HH;
        const float* w = wq + a * HH;
        float acc = bq[a];
        for (int k = 0; k < HH; ++k) acc += hrow[k] * w[k];
        qg[(b0 + bb) * 16 + a] = acc;
    }
}

// -------- attention: conv(prev attn) + tanh + softmax(T) + context --------
__global__ void k_attn(int step,
                       const float* __restrict__ enc, const float* __restrict__ vproj,
                       const float* __restrict__ qg,
                       const float* __restrict__ cw, const float* __restrict__ cb,
                       const float* __restrict__ abias,
                       const float* __restrict__ sw, const float* __restrict__ sbp,
                       float* __restrict__ aprevg, float* __restrict__ ctxg,
                       float* __restrict__ attn_out) {
    __shared__ float ap[TT];
    __shared__ float sc[TT];
    __shared__ float red[512];
    __shared__ float cpart[4 * VDIM];
    __shared__ float cws[ADIM * 3], cbs[ADIM], qv[ADIM], sws[ADIM];
    int tid = threadIdx.x;
    int b = blockIdx.x;
    if (tid < ADIM * 3) cws[tid] = cw[tid];
    if (tid < ADIM) {
        cbs[tid] = cb[tid];
        qv[tid]  = qg[b * 16 + tid] + abias[tid];
        sws[tid] = sw[tid];
    }
    for (int t = tid; t < TT; t += 512) ap[t] = aprevg[b * TT + t];
    __syncthreads();
    float sb = sbp[0];
    for (int t = tid; t < TT; t += 512) {
        float am1 = (t > 0)      ? ap[t - 1] : 0.f;
        float a0  = ap[t];
        float ap1 = (t < TT - 1) ? ap[t + 1] : 0.f;
        const float* vp = vproj + ((long)b * TT + t) * 16;
        float s = sb;
#pragma unroll
        for (int a = 0; a < ADIM; ++a) {
            float loc = cws[a * 3] * am1 + cws[a * 3 + 1] * a0 + cws[a * 3 + 2] * ap1 + cbs[a];
            s += tanhf(qv[a] + vp[a] + loc) * sws[a];
        }
        sc[t] = s;
    }
    __syncthreads();
    float lm = -1e30f;
    for (int t = tid; t < TT; t += 512) lm = fmaxf(lm, sc[t]);
    red[tid] = lm; __syncthreads();
    for (int off = 256; off > 0; off >>= 1) {
        if (tid < off) red[tid] = fmaxf(red[tid], red[tid + off]);
        __syncthreads();
    }
    float mx = red[0]; __syncthreads();
    float ls = 0.f;
    for (int t = tid; t < TT; t += 512) { float e = expf(sc[t] - mx); sc[t] = e; ls += e; }
    red[tid] = ls; __syncthreads();
    for (int off = 256; off > 0; off >>= 1) {
        if (tid < off) red[tid] += red[tid + off];
        __syncthreads();
    }
    float inv = 1.f / red[0];
    __syncthreads();
    for (int t = tid; t < TT; t += 512) {
        float v = sc[t] * inv;
        sc[t] = v;
        aprevg[b * TT + t] = v;
        attn_out[((long)b * STEPS + step) * TT + t] = v;
    }
    __syncthreads();
    // context[b][v] = sum_t attn[t] * enc[b][t][v]  (v-contiguous -> coalesced)
    int v  = tid & (VDIM - 1);
    int tg = tid >> 7;
    float acc = 0.f;
    for (int t = tg; t < TT; t += 4) acc += sc[t] * enc[((long)b * TT + t) * VDIM + v];
    cpart[tg * VDIM + v] = acc;
    __syncthreads();
    if (tid < VDIM) {
        ctxg[b * VDIM + tid] = cpart[tid] + cpart[VDIM + tid] +
                               cpart[2 * VDIM + tid] + cpart[3 * VDIM + tid];
    }
}

// -------- classifier: fc1+fc2 (VALU), fc3 via f32 WMMA (16b x 4000 x 48) --------
__global__ void k_fc(const float* __restrict__ h1g, const float* __restrict__ ctxg,
                     const float* __restrict__ w1, const float* __restrict__ b1,
                     const float* __restrict__ w2, const float* __restrict__ b2,
                     const float* __restrict__ w3, const float* __restrict__ b3,
                     float* __restrict__ logits) {
    __shared__ float hcs[16 * 192];
    __shared__ float y1s[16 * HS1];
    __shared__ float y2s[16 * HS2];
    int tid = threadIdx.x;
    int b0  = blockIdx.x * 16;
    for (int i = tid; i < 16 * 192; i += 256) {
        int bb = i / 192, k = i % 192;
        hcs[i] = (k < HH) ? h1g[(b0 + bb) * HH + k] : ctxg[(b0 + bb) * VDIM + (k - HH)];
    }
    __syncthreads();
    for (int i = tid; i < 16 * HS1; i += 256) {
        int bb = i / HS1, j = i % HS1;
        const float* w = w1 + j * 192;
        const float* hc = hcs + bb * 192;
        float acc = b1[j];
        for (int k = 0; k < 192; ++k) acc += hc[k] * w[k];
        y1s[i] = fmaxf(acc, 0.f);
    }
    __syncthreads();
    for (int i = tid; i < 16 * HS2; i += 256) {
        int bb = i / HS2, j = i % HS2;
        const float* w = w2 + j * HS1;
        const float* y = y1s + bb * HS1;
        float acc = b2[j];
        for (int k = 0; k < HS1; ++k) acc += y[k] * w[k];
        y2s[i] = fmaxf(acc, 0.f);
    }
    __syncthreads();
    int wave = tid >> 5, lane = tid & 31;
    int m  = lane & 15;
    int kh = (lane >> 4) * 2;
    int mr = (lane >> 4) * 8;
    for (int nt = wave; nt < NC / 16; nt += 8) {   // uniform per wave -> EXEC stays full
        int gn = nt * 16 + (lane & 15);
        float bias = b3[gn];
        const float* brow = w3 + gn * HS2 + kh;
        const float* arow = y2s + m * HS2 + kh;
        v8f acc = {};
#pragma unroll
        for (int k0 = 0; k0 < HS2; k0 += 4) {
            v2f a;  a.x = arow[k0];  a.y = arow[k0 + 1];
            v2f bf; bf.x = brow[k0]; bf.y = brow[k0 + 1];
            acc = __builtin_amdgcn_wmma_f32_16x16x4_f32(
                false, a, false, bf, (short)0, acc, false, false);
        }
        float* dst = logits + (long)(b0 + mr) * NC + gn;
#pragma unroll
        for (int r = 0; r < 8; ++r) dst[(long)r * NC] = acc[r] + bias;
    }
}

// -------- log_softmax over 4000 classes per batch row --------
__global__ void k_lsm(int step, const float* __restrict__ logits,
                      float* __restrict__ out) {
    __shared__ float red[256];
    int tid = threadIdx.x, b = blockIdx.x;
    const float* row = logits + (long)b * NC;
    float lm = -1e30f;
    for (int c = tid; c < NC; c += 256) lm = fmaxf(lm, row[c]);
    red[tid] = lm; __syncthreads();
    for (int off = 128; off > 0; off >>= 1) {
        if (tid < off) red[tid] = fmaxf(red[tid], red[tid + off]);
        __syncthreads();
    }
    float mx = red[0]; __syncthreads();
    float ls = 0.f;
    for (int c = tid; c < NC; c += 256) ls += expf(row[c] - mx);
    red[tid] = ls; __syncthreads();
    for (int off = 128; off > 0; off >>= 1) {
        if (tid < off) red[tid] += red[tid + off];
        __syncthreads();
    }
    float lse = mx + logf(red[0]);
    float* o = out + ((long)b * STEPS + step) * NC;
    for (int c = tid; c < NC; c += 256) o[c] = row[c] - lse;
}

extern "C" void kernel_launch(void* const* d_in, const int* in_sizes, int n_in,
                              void* d_out, int out_size, void* d_ws, size_t ws_size,
                              hipStream_t stream) {
    const float* enc   = (const float*)d_in[0];
    const float* whh0  = (const float*)d_in[2];
    const float* bih0  = (const float*)d_in[3];
    const float* bhh0  = (const float*)d_in[4];
    const float* wih1  = (const float*)d_in[5];
    const float* whh1  = (const float*)d_in[6];
    const float* bih1  = (const float*)d_in[7];
    const float* bhh1  = (const float*)d_in[8];
    const float* cw    = (const float*)d_in[9];
    const float* cb    = (const float*)d_in[10];
    const float* wq    = (const float*)d_in[11];
    const float* bq    = (const float*)d_in[12];
    const float* wv    = (const float*)d_in[13];
    const float* abias = (const float*)d_in[14];
    const float* sw    = (const float*)d_in[15];
    const float* sb    = (const float*)d_in[16];
    const float* w1    = (const float*)d_in[17];
    const float* b1    = (const float*)d_in[18];
    const float* w2    = (const float*)d_in[19];
    const float* b2    = (const float*)d_in[20];
    const float* w3    = (const float*)d_in[21];
    const float* b3    = (const float*)d_in[22];

    float* ws     = (float*)d_ws;
    float* vproj  = ws;                                   // B*T*16
    float* h0g    = vproj + (size_t)BB * TT * 16;         // B*H
    float* h1g    = h0g + BB * HH;                        // B*H
    float* qg     = h1g + BB * HH;                        // B*16
    float* aprev  = qg + BB * 16;                         // B*T
    float* ctx    = aprev + (size_t)BB * TT;              // B*128
    float* logits = ctx + BB * VDIM;                      // B*4000

    float* out_logp = (float*)d_out;
    float* out_attn = out_logp + (size_t)BB * STEPS * NC;

    k_init<<<(BB * TT + 255) / 256, 256, 0, stream>>>(h0g, h1g, aprev);
    k_vproj<<<(BB * TT / 16) / 8, 256, 0, stream>>>(enc, wv, vproj);

    for (int step = 0; step < STEPS; ++step) {
        k_gru<<<BB / 16, 256, 0, stream>>>(whh0, bih0, bhh0, wih1, whh1,
                                           bih1, bhh1, wq, bq, h0g, h1g, qg);
        k_attn<<<BB, 512, 0, stream>>>(step, enc, vproj, qg, cw, cb, abias,
                                       sw, sb, aprev, ctx, out_attn);
        k_fc<<<BB / 16, 256, 0, stream>>>(h1g, ctx, w1, b1, w2, b2, w3, b3, logits);
        k_lsm<<<BB, 256, 0, stream>>>(step, logits, out_logp);
    }
    (void)in_sizes; (void)n_in; (void)out_size; (void)ws_size;
}